// HadamardEdgeEmb_34256659153222
// MI455X (gfx1250) — compile-verified
//
#include <hip/hip_runtime.h>

typedef __attribute__((ext_vector_type(2))) float v2f;
typedef __attribute__((ext_vector_type(8))) float v8f;

#define D_FEAT     128
#define LDS_STRIDE 132   // 128 + 4 pad: lane bank = (4*row + col) % 64 -> conflict-free frag reads
#define TILE_M     16

__global__ __launch_bounds__(256) void edge_gemm_relu_kernel(
    const float* __restrict__ nodes,   // [N, 128]
    const int*   __restrict__ eidx,    // [2, E]
    const float* __restrict__ W,       // [128, 128] row-major (out, in)
    const float* __restrict__ bias,    // [128]
    float*       __restrict__ out,     // [E, 128]
    int E, int numTiles)
{
    __shared__ float Atile[2][TILE_M * LDS_STRIDE];

    const int tid  = threadIdx.x;
    const int lane = tid & 31;
    const int wave = tid >> 5;       // 0..7 -> output-column tile
    const int nBase = wave * 16;
    const int half  = lane >> 4;     // 0: K even pair, 1: K odd pair (A/B frag layout)
    const int l16   = lane & 15;

    // ---- Preload B fragments (whole of W^T for this wave's 16 columns) into VGPRs.
    // B[k'][n] = W[n][k'] ; step k: lane<16 -> W[nBase+l16][4k+0..1], lane>=16 -> [4k+2..3]
    v2f bfrag[32];
    {
        const float* wrow = W + (nBase + l16) * D_FEAT + 2 * half;
        #pragma unroll
        for (int k = 0; k < 32; ++k)
            bfrag[k] = *(const v2f*)(wrow + 4 * k);
    }
    const float bval = bias[nBase + l16];

    const int* srcIdx = eidx;        // row 0 of edge_index
    const int* dstIdx = eidx + E;    // row 1

    // Cooperative-fill mapping: 16 threads per edge, 8 floats per thread.
    const int fe = tid >> 4;         // edge-in-tile 0..15
    const int f0 = (tid & 15) * 8;   // feature start

    // Gather prefetch registers (raw; summed only at LDS-deposit time so the
    // loadcnt wait is deferred across the compute of the previous tile).
    float4 s0, s1, d0, d1;

    auto gather = [&](int t) {
        int e = t * TILE_M + fe;
        if (e >= E) e = E - 1;            // clamp tail; stores masked later
        const int si = srcIdx[e];
        const int di = dstIdx[e];
        const float4* sp = (const float4*)(nodes + (size_t)si * D_FEAT + f0);
        const float4* dp = (const float4*)(nodes + (size_t)di * D_FEAT + f0);
        s0 = sp[0]; s1 = sp[1];
        d0 = dp[0]; d1 = dp[1];
    };

    int t = blockIdx.x;
    if (t < numTiles) gather(t);         // uniform guard (blockIdx-only)
    int buf = 0;

    for (; t < numTiles; t += gridDim.x) {
        // ---- Deposit current tile's 16x128 A-tile into the Atile LDS buffer selected by buf ----
        {
            float4* lp = (float4*)&Atile[buf][fe * LDS_STRIDE + f0];
            lp[0] = make_float4(s0.x + d0.x, s0.y + d0.y, s0.z + d0.z, s0.w + d0.w);
            lp[1] = make_float4(s1.x + d1.x, s1.y + d1.y, s1.z + d1.z, s1.w + d1.w);
        }
        __syncthreads();   // buf writes visible; also fences reads of buf from 2 iters ago

        // ---- Kick off next tile's gather; latency hidden under the WMMAs below ----
        const int tn = t + gridDim.x;
        if (tn < numTiles) gather(tn);   // uniform branch

        // ---- 16x16 output tile: 32 v_wmma_f32_16x16x4_f32 over K=128,
        //      two independent accumulator chains for ILP ----
        v8f acc0 = {0.f, 0.f, 0.f, 0.f, 0.f, 0.f, 0.f, 0.f};
        v8f acc1 = {0.f, 0.f, 0.f, 0.f, 0.f, 0.f, 0.f, 0.f};
        const float* arow = &Atile[buf][l16 * LDS_STRIDE + 2 * half];
        #pragma unroll
        for (int k = 0; k < 32; k += 2) {
            v2f a0 = *(const v2f*)(arow + 4 * k);        // ds_load_b64, conflict-free
            v2f a1 = *(const v2f*)(arow + 4 * k + 4);
            acc0 = __builtin_amdgcn_wmma_f32_16x16x4_f32(
                       false, a0, false, bfrag[k],     (short)0, acc0, false, false);
            acc1 = __builtin_amdgcn_wmma_f32_16x16x4_f32(
                       false, a1, false, bfrag[k + 1], (short)0, acc1, false, false);
        }

        // ---- bias + ReLU + non-temporal store (keep node table resident in L2) ----
        const int e0 = t * TILE_M;
        #pragma unroll
        for (int v = 0; v < 8; ++v) {
            const int m = v + 8 * half;      // row within tile (C/D VGPR layout)
            const int e = e0 + m;
            if (e < E) {
                float r = acc0[v] + acc1[v] + bval;
                r = r > 0.f ? r : 0.f;
                __builtin_nontemporal_store(r, out + (size_t)e * D_FEAT + nBase + l16);
            }
        }

        buf ^= 1;
    }
}

extern "C" void kernel_launch(void* const* d_in, const int* in_sizes, int n_in,
                              void* d_out, int out_size, void* d_ws, size_t ws_size,
                              hipStream_t stream) {
    const float* nodes = (const float*)d_in[0];
    const int*   eidx  = (const int*)d_in[1];   // int32 per harness convention
    const float* W     = (const float*)d_in[2];
    const float* bias  = (const float*)d_in[3];
    float*       out   = (float*)d_out;

    const int E        = in_sizes[1] / 2;
    const int numTiles = (E + TILE_M - 1) / TILE_M;
    int blocks = numTiles < 2048 ? numTiles : 2048;
    if (blocks < 1) blocks = 1;

    hipLaunchKernelGGL(edge_gemm_relu_kernel, dim3(blocks), dim3(256), 0, stream,
                       nodes, eidx, W, bias, out, E, numTiles);
}